// SelectiveScan_59760174957209
// MI455X (gfx1250) — compile-verified
//
#include <hip/hip_runtime.h>

// CDNA5 / gfx1250, wave32. Selective-scan (SSM) via chunked linear scan in
// TRANSPOSED form: per channel m, per chunk of T=16 timesteps,
//   Y^T     = W^T @ X^T + Chat^T @ S^T          (D-skip folded into W diag)
//   S_new^T = Bhat^T @ X^T + Ahat^T @ S^T
// All products are V_WMMA_F32_16X16X4_F32 (K=16 as 4 chained K=4 slices).
// Constant per-channel matrices are the A-operands; data (X^T, S^T) are the
// B-operands. S^T stays in WMMA D-layout across chunks; re-shaping it into
// the next chunk's B-operand needs only a lane-half swap, done with
// v_permlanex16 (pure VALU) to keep the serialized scan chain off LDS.

typedef __attribute__((ext_vector_type(2))) float v2f;
typedef __attribute__((ext_vector_type(8))) float v8f;

#define D_MODEL 1024
#define D_STATE 16
#define BATCH   8
#define SEQ_LEN 4096
#define CHUNK   16
#define NCHUNK  (SEQ_LEN / CHUNK)

// Workspace per channel (1024 floats), all matrices stored TRANSPOSED,
// row-major [M row][K col] so A-operand slices are contiguous v2f pairs:
//   [  0..255] Wt : Wt[j][i] = k_{j-i} (i<=j else 0), k_0 += Dv
//   [256..511] Ct : Ct[j][s] = (A^{j+1} Cv^T)[s]
//   [512..767] Bt : Bt[t][i] = (Bv A^{15-i})[t]
//   [768..1023] At: At[t][s] = A^16[s][t]

__global__ __launch_bounds__(256) void ssm_precompute(
    const float* __restrict__ A, const float* __restrict__ B,
    const float* __restrict__ C, const float* __restrict__ Dp,
    float* __restrict__ ws)
{
    __shared__ float Alds[256];
    __shared__ float P[256];
    __shared__ float Pn[256];
    __shared__ float Bv[16], Cv[16], karr[16], bh[16];

    const int m   = blockIdx.x;
    const int tid = threadIdx.x;
    const int i   = tid >> 4;
    const int j   = tid & 15;

    Alds[tid] = A[(size_t)m * 256 + tid];
    if (tid < 16) { Bv[tid] = B[(size_t)m * 16 + tid]; Cv[tid] = C[(size_t)m * 16 + tid]; }
    P[tid] = (i == j) ? 1.0f : 0.0f;   // P = A^0 = I
    __syncthreads();

    float*      wsb = ws + (size_t)m * 1024;
    const float Dv  = Dp[m];

    for (int d = 0; d <= 16; ++d) {
        // bh[t] = sum_s Bv[s] * P[s][t]  (P = A^d);  Bt column (15-d)
        if (tid < 16) {
            float acc = 0.0f;
            #pragma unroll
            for (int s = 0; s < 16; ++s) acc += Bv[s] * P[s * 16 + tid];
            bh[tid] = acc;
            if (d <= 15) wsb[512 + tid * 16 + (15 - d)] = acc;   // Bt[t][15-d]
        }
        // Ct row (d-1): Ct[d-1][s] = sum_t P[s][t] * Cv[t]
        if (tid >= 16 && tid < 32 && d >= 1) {
            const int s = tid - 16;
            float acc = 0.0f;
            #pragma unroll
            for (int t = 0; t < 16; ++t) acc += P[s * 16 + t] * Cv[t];
            wsb[256 + (d - 1) * 16 + s] = acc;
        }
        __syncthreads();
        // k_d = Bv A^d Cv^T
        if (tid == 0 && d <= 15) {
            float acc = 0.0f;
            #pragma unroll
            for (int t = 0; t < 16; ++t) acc += bh[t] * Cv[t];
            karr[d] = acc;
        }
        if (d == 16) wsb[768 + j * 16 + i] = P[i * 16 + j];   // At = (A^16)^T
        if (d < 16) {                                          // Pn = P @ A
            float acc = 0.0f;
            #pragma unroll
            for (int k = 0; k < 16; ++k) acc += P[i * 16 + k] * Alds[k * 16 + j];
            Pn[tid] = acc;
        }
        __syncthreads();
        if (d < 16) P[tid] = Pn[tid];
        __syncthreads();
    }
    // Wt (lower-triangular Toeplitz of k_d, transposed form), D folded on diag
    float wv = (j <= i) ? karr[i - j] : 0.0f;
    if (i == j) wv += Dv;
    wsb[tid] = wv;
}

// Lane-half swap (lane l <-> lane l^16) as a VALU op: v_permlanex16 with
// identity nibble selects. Passing `v` as the tied "old" operand avoids a
// dead v_mov 0 before each permlane (old is never selected: fi=bc=false,
// all lanes active).
__device__ __forceinline__ float lane_half_swap(float v) {
    unsigned s = __float_as_uint(v);
    unsigned r = __builtin_amdgcn_permlanex16(
        s, s, 0x76543210u, 0xfedcba98u, false, false);
    return __uint_as_float(r);
}

// One wave (32 threads) per channel. B-operand columns carry the 8 batches
// (cols 8..15 unused). No LDS: state is carried in the WMMA D accumulator
// and re-shaped via a VALU lane-half swap each chunk.
__global__ __launch_bounds__(32) void ssm_scan(
    const float* __restrict__ x, const float* __restrict__ ws,
    float* __restrict__ y)
{
    const int m  = blockIdx.x;
    const int l  = threadIdx.x;
    const int n  = l & 15;     // column (batch) for B/D; M row for A-operand
    const int hi = l >> 4;     // lane half

    const float* wsb = ws + (size_t)m * 1024;

    // Constant A-operand fragments: A-op 16x4 f32 layout (ISA 7.12.2):
    // lane l -> row l%16; lanes 0-15 hold K={k0,k0+1}, lanes 16-31 K={k0+2,k0+3}.
    // Matrix col pair (4*sl + 2*hi, +1) is contiguous -> one v2f load each.
    v2f Wt[4], Ct[4], Bt[4], At[4];
    const v2f* wp = (const v2f*)wsb;
    #pragma unroll
    for (int sl = 0; sl < 4; ++sl) {
        const int idx = n * 8 + sl * 2 + hi;
        Wt[sl] = wp[      idx];
        Ct[sl] = wp[128 + idx];
        Bt[sl] = wp[256 + idx];
        At[sl] = wp[384 + idx];
    }

    // Per-lane streaming pointers; all per-chunk accesses use imm offsets.
    const int    bn = n & 7;
    const float* xp = x + (size_t)bn * SEQ_LEN * D_MODEL + m;
    float*       yp = y + (size_t)bn * SEQ_LEN * D_MODEL + m;

    v8f sacc = {};             // S^T in D-layout (rows=state, cols=batch)
    v2f Xf[4];
    #pragma unroll
    for (int sl = 0; sl < 4; ++sl) { v2f z; z.x = 0.0f; z.y = 0.0f; Xf[sl] = z; }

    #pragma unroll 2
    for (int c = 0; c < NCHUNK; ++c) {
        // X^T B-operand: lane (col=n=batch, half=hi) holds time rows
        // k = 4*sl + 2*hi {+1}. Cols 8..15 keep zeros.
        if (n < BATCH) {
            #pragma unroll
            for (int sl = 0; sl < 4; ++sl) {
                const int k0 = 4 * sl + 2 * hi;
                v2f t;
                t.x = xp[(size_t)(k0    ) * D_MODEL];
                t.y = xp[(size_t)(k0 + 1) * D_MODEL];
                Xf[sl] = t;
            }
        }

        // S^T B-operand from D-layout: own regs + partner lane-half regs.
        float pv[8];
        #pragma unroll
        for (int v = 0; v < 8; ++v) pv[v] = lane_half_swap((float)sacc[v]);
        v2f Sf[4];
        {
            v2f s0, s1, s2, s3;
            s0.x = (hi == 0) ? (float)sacc[0] : pv[2];
            s0.y = (hi == 0) ? (float)sacc[1] : pv[3];
            s1.x = (hi == 0) ? (float)sacc[4] : pv[6];
            s1.y = (hi == 0) ? (float)sacc[5] : pv[7];
            s2.x = (hi == 0) ? pv[0] : (float)sacc[2];
            s2.y = (hi == 0) ? pv[1] : (float)sacc[3];
            s3.x = (hi == 0) ? pv[4] : (float)sacc[6];
            s3.y = (hi == 0) ? pv[5] : (float)sacc[7];
            Sf[0] = s0; Sf[1] = s1; Sf[2] = s2; Sf[3] = s3;
        }

        // Y^T = W^T @ X^T + Chat^T @ S^T   (x*Dv lives on W^T's diagonal)
        v8f yacc = {};
        #pragma unroll
        for (int sl = 0; sl < 4; ++sl)
            yacc = __builtin_amdgcn_wmma_f32_16x16x4_f32(
                false, Wt[sl], false, Xf[sl], (short)0, yacc, false, false);
        #pragma unroll
        for (int sl = 0; sl < 4; ++sl)
            yacc = __builtin_amdgcn_wmma_f32_16x16x4_f32(
                false, Ct[sl], false, Sf[sl], (short)0, yacc, false, false);

        // S_new^T = Bhat^T @ X^T (independent, overlaps the swap) then
        //           Ahat^T @ S^T (the only cross-chunk dependency, last)
        v8f ns = {};
        #pragma unroll
        for (int sl = 0; sl < 4; ++sl)
            ns = __builtin_amdgcn_wmma_f32_16x16x4_f32(
                false, Bt[sl], false, Xf[sl], (short)0, ns, false, false);
        #pragma unroll
        for (int sl = 0; sl < 4; ++sl)
            ns = __builtin_amdgcn_wmma_f32_16x16x4_f32(
                false, At[sl], false, Sf[sl], (short)0, ns, false, false);
        sacc = ns;

        // Y^T store: vgpr v -> time row (v + 8*hi), col n = batch.
        if (n < BATCH) {
            #pragma unroll
            for (int v = 0; v < 8; ++v)
                yp[(size_t)(v + 8 * hi) * D_MODEL] = yacc[v];
        }

        xp += CHUNK * D_MODEL;
        yp += CHUNK * D_MODEL;
    }
}

extern "C" void kernel_launch(void* const* d_in, const int* in_sizes, int n_in,
                              void* d_out, int out_size, void* d_ws, size_t ws_size,
                              hipStream_t stream) {
    const float* x  = (const float*)d_in[0];
    const float* A  = (const float*)d_in[1];
    const float* B  = (const float*)d_in[2];
    const float* C  = (const float*)d_in[3];
    const float* D  = (const float*)d_in[4];
    float*       ws = (float*)d_ws;    // 1024 channels * 1024 floats = 4 MB
    float*       y  = (float*)d_out;

    ssm_precompute<<<D_MODEL, 256, 0, stream>>>(A, B, C, D, ws);
    ssm_scan<<<D_MODEL, 32, 0, stream>>>(x, ws, y);
}